// NonMaxSuppression_41532333752560
// MI455X (gfx1250) — compile-verified
//
#include <hip/hip_runtime.h>
#include <stdint.h>

// NMS for (B=4, N=2048, 16 classes) on gfx1250.
// Latency/L2-bound problem: stage each image tile into LDS via the CDNA5
// Tensor Data Mover, sort + greedy-suppress in LDS with wave32 blocks.

#define NUM_CLASSES 16
#define NBOX        2048
#define IOU_TH      0.5f
#define SCORE_TH    0.05f
#define MAX_DET     100
#define MAX_PER_CLASS 100
#define NT          256           // 8 waves (wave32)
#define PRED_DW     (NBOX * 6)    // 12288 dwords per image
#define CHUNK_DW    (PRED_DW / 8) // 1536 dwords per wave

typedef unsigned int u32;
typedef u32 u32x4 __attribute__((ext_vector_type(4)));
typedef int i32x4 __attribute__((ext_vector_type(4)));
typedef int i32x8 __attribute__((ext_vector_type(8)));

// ---------------------------------------------------------------------------
// TDM: DMA a contiguous CHUNK_DW-dword chunk of global memory into LDS.
// Descriptor packing per CDNA5 ISA (group0) / §8.4
// (group1). 1-D tile: tensor_dim0 = tile_dim0 = tensor_dim0_stride = 1536,
// data_size = 4 bytes, no padding, no multicast (not in a cluster).
// ---------------------------------------------------------------------------
__device__ inline void tdm_load_chunk_to_lds(u32 lds_byte_addr, const void* gsrc) {
  uint64_t ga = (uint64_t)(uintptr_t)gsrc;

  u32x4 g0;
  g0[0] = 1u;                                        // count=1 (valid), is_restore=0, gather off
  g0[1] = lds_byte_addr;                             // LDS destination byte address
  g0[2] = (u32)(ga & 0xffffffffu);                   // global_addr[31:0]
  g0[3] = (u32)((ga >> 32) & 0x01ffffffu)            // global_addr[56:32]
        | (2u << 30);                                // type = 2 ("image")

  i32x8 g1;
  g1[0] = (2 << 16);            // workgroup_mask=0, data_size=2 (4 bytes), flags=0
  g1[1] = (CHUNK_DW << 16);     // atomic_barrier_addr=0 | tensor_dim0[15:0]
  g1[2] = (1 << 16);            // tensor_dim0[31:16]=0 | tensor_dim1[15:0]=1
  g1[3] = (CHUNK_DW << 16);     // tensor_dim1[31:16]=0 | tile_dim0=1536
  g1[4] = 1;                    // tile_dim1=1 | tile_dim2=0
  g1[5] = CHUNK_DW;             // tensor_dim0_stride[31:0]
  g1[6] = 0;                    // tensor_dim0_stride[47:32]=0 | tensor_dim1_stride[15:0]=0
  g1[7] = 0;                    // tensor_dim1_stride[47:16]=0

  i32x4 z4 = {0, 0, 0, 0};
#if __has_include(<hip/amd_detail/amd_gfx1250_TDM.h>)
  // therock / clang-23 toolchain: 6-arg form
  i32x8 z8 = {0, 0, 0, 0, 0, 0, 0, 0};
  __builtin_amdgcn_tensor_load_to_lds(g0, g1, z4, z4, z8, 0);
#else
  // ROCm 7.2 / clang-22 toolchain: 5-arg form
  __builtin_amdgcn_tensor_load_to_lds(g0, g1, z4, z4, 0);
#endif
}

// ---------------------------------------------------------------------------
// In-LDS bitonic sort, descending by key, carrying an int payload.
// NN must be a power of two; NT threads cooperate; NN/2 compare-exchanges
// per stage, disjoint pairs per thread, barrier between stages.
// ---------------------------------------------------------------------------
template <int NN>
__device__ inline void bitonic_sort_desc(float* key, int* val, int tid) {
  for (int k = 2; k <= NN; k <<= 1) {
    for (int j = k >> 1; j > 0; j >>= 1) {
      __syncthreads();
      for (int t = tid; t < NN / 2; t += NT) {
        int i = ((t & ~(j - 1)) << 1) | (t & (j - 1)); // bit log2(j) of i is 0
        int p = i | j;
        bool desc = ((i & k) == 0);
        float a = key[i], b = key[p];
        bool sw = desc ? (a < b) : (a > b);
        if (sw) {
          key[i] = b; key[p] = a;
          int tv = val[i]; val[i] = val[p]; val[p] = tv;
        }
      }
    }
  }
  __syncthreads();
}

__device__ inline float iou_box(const float* a, const float* b) {
  float areaA = (a[2] - a[0]) * (a[3] - a[1]);
  float areaB = (b[2] - b[0]) * (b[3] - b[1]);
  float iw = fmaxf(fminf(a[2], b[2]) - fmaxf(a[0], b[0]), 0.0f);
  float ih = fmaxf(fminf(a[3], b[3]) - fmaxf(a[1], b[1]), 0.0f);
  float inter = iw * ih;
  float uni = areaA + areaB - inter;
  return inter > 0.0f ? inter / fmaxf(uni, 1e-8f) : 0.0f;
}

// ---------------------------------------------------------------------------
// Kernel 1: one block per (image b, class c).
//   TDM-stage image -> mask+sort scores -> greedy NMS -> per-class top-100
//   compacted into workspace: entry stride 8 floats:
//   [x1,y1,x2,y2, score, class, 0, 0], flat index (b*16+c)*100 + slot.
// ---------------------------------------------------------------------------
__global__ void __launch_bounds__(NT)
nms_per_class_kernel(const float* __restrict__ pred_g, float* __restrict__ S) {
  __shared__ float pred[PRED_DW];   // 48 KB image tile
  __shared__ float skey[NBOX];      // sort keys (masked scores)
  __shared__ int   sidx[NBOX];      // payload: original box index
  __shared__ int   keepf[NBOX];     // keep flags over sorted order
  __shared__ int   sM;

  const int tid = threadIdx.x;
  const int b   = blockIdx.x / NUM_CLASSES;
  const int c   = blockIdx.x % NUM_CLASSES;

  // --- Stage: 8 waves each DMA one 6KB chunk of the image into LDS (TDM) ---
  {
    const int wave = tid >> 5;
    u32 lds_base = (u32)(uintptr_t)(&pred[0]);     // LDS byte offset (low 32 bits)
    const float* gsrc = pred_g + (size_t)b * PRED_DW + (size_t)wave * CHUNK_DW;
    tdm_load_chunk_to_lds(lds_base + (u32)wave * (u32)CHUNK_DW * 4u, (const void*)gsrc);
    __builtin_amdgcn_s_wait_tensorcnt(0);
  }
  if (tid == 0) sM = 0;
  __syncthreads();

  // --- Build masked keys: score iff (class == c && score > SCORE_TH) ---
  for (int i = tid; i < NBOX; i += NT) {
    float cls = pred[i * 6 + 4];
    float s   = pred[i * 6 + 5];
    bool cand = ((int)cls == c) && (s > SCORE_TH);
    skey[i] = cand ? s : -1.0f;
    sidx[i] = i;
  }

  // --- Sort descending by score (candidates first, score order) ---
  bitonic_sort_desc<NBOX>(skey, sidx, tid);

  // --- Count candidates M ---
  {
    int local = 0;
    for (int i = tid; i < NBOX; i += NT)
      if (skey[i] > SCORE_TH) local++;
    if (local) atomicAdd(&sM, local);
  }
  __syncthreads();
  const int M = sM;

  for (int i = tid; i < NBOX; i += NT) keepf[i] = (i < M) ? 1 : 0;

  // --- Greedy suppression: sequential over kept i, parallel over j > i ---
  for (int i = 0; i < M; ++i) {
    __syncthreads();                   // make prior keepf writes visible
    if (!keepf[i]) continue;           // uniform branch (same LDS word read by all)
    const float* bp = &pred[sidx[i] * 6];
    float bi[4] = {bp[0], bp[1], bp[2], bp[3]};
    for (int jj = i + 1 + tid; jj < M; jj += NT) {
      if (keepf[jj]) {
        if (iou_box(bi, &pred[sidx[jj] * 6]) > IOU_TH) keepf[jj] = 0;
      }
    }
  }
  __syncthreads();

  // --- Compact first MAX_PER_CLASS kept detections into workspace ---
  if (tid == 0) {
    float* base = S + (size_t)(b * NUM_CLASSES + c) * MAX_PER_CLASS * 8;
    int cnt = 0;
    for (int r = 0; r < M && cnt < MAX_PER_CLASS; ++r) {
      if (keepf[r]) {
        const float* bp = &pred[sidx[r] * 6];
        float* e = base + (size_t)cnt * 8;
        e[0] = bp[0]; e[1] = bp[1]; e[2] = bp[2]; e[3] = bp[3];
        e[4] = skey[r]; e[5] = (float)c; e[6] = 0.0f; e[7] = 0.0f;
        ++cnt;
      }
    }
    for (; cnt < MAX_PER_CLASS; ++cnt) {
      float* e = base + (size_t)cnt * 8;
      e[0] = e[1] = e[2] = e[3] = e[4] = e[5] = e[6] = e[7] = 0.0f;
    }
  }
}

// ---------------------------------------------------------------------------
// Kernel 2: one block per image. Top-100 over the 1600 per-class entries,
// emit [x1,y1,x2,y2,class,score] rows + nvalid (as float) at the tail.
// ---------------------------------------------------------------------------
__global__ void __launch_bounds__(NT)
nms_final_topk_kernel(const float* __restrict__ S, float* __restrict__ out, int B) {
  __shared__ float key[2048];
  __shared__ int   val[2048];
  __shared__ int   cntS;

  const int tid = threadIdx.x;
  const int b   = blockIdx.x;
  const int TOT = NUM_CLASSES * MAX_PER_CLASS; // 1600

  if (tid == 0) cntS = 0;
  for (int i = tid; i < 2048; i += NT) {
    if (i < TOT) {
      key[i] = S[((size_t)b * TOT + i) * 8 + 4];  // stored score (0 if invalid slot)
      val[i] = i;
    } else {
      key[i] = -2.0f;
      val[i] = -1;
    }
  }

  bitonic_sort_desc<2048>(key, val, tid);

  if (tid < MAX_DET) {
    float s = key[tid];
    int   j = val[tid];
    float r0 = 0.f, r1 = 0.f, r2 = 0.f, r3 = 0.f, r4 = 0.f, r5 = 0.f;
    if (s > SCORE_TH && j >= 0) {
      const float* e = S + ((size_t)b * TOT + j) * 8;
      r0 = e[0]; r1 = e[1]; r2 = e[2]; r3 = e[3];
      r4 = e[5];             // class
      r5 = s;                // score
      atomicAdd(&cntS, 1);
    }
    float* o = out + ((size_t)b * MAX_DET + tid) * 6;
    o[0] = r0; o[1] = r1; o[2] = r2; o[3] = r3; o[4] = r4; o[5] = r5;
  }
  __syncthreads();
  if (tid == 0) out[(size_t)B * MAX_DET * 6 + b] = (float)cntS;
}

// ---------------------------------------------------------------------------
extern "C" void kernel_launch(void* const* d_in, const int* in_sizes, int n_in,
                              void* d_out, int out_size, void* d_ws, size_t ws_size,
                              hipStream_t stream) {
  const float* pred = (const float*)d_in[0];
  float* out = (float*)d_out;
  float* S   = (float*)d_ws;          // B*16*100*8 floats = 200 KB for B=4
  const int B = in_sizes[0] / (NBOX * 6);

  nms_per_class_kernel<<<B * NUM_CLASSES, NT, 0, stream>>>(pred, S);
  nms_final_topk_kernel<<<B, NT, 0, stream>>>(S, out, B);
}